// KGCN_33122787786944
// MI455X (gfx1250) — compile-verified
//
#include <hip/hip_runtime.h>
#include <hip/hip_bf16.h>

#define BATCH 65536
#define NNBR 8
#define DIM 16

typedef __attribute__((ext_vector_type(2))) float v2f;
typedef __attribute__((ext_vector_type(8))) float v8f;

// ---------------------------------------------------------------------------
// ws[0] = sum of BCE losses, ws[1] = sum of squares (all three tables)
// ---------------------------------------------------------------------------
__global__ void zero_ws_kernel(float* ws) {
    ws[0] = 0.0f;
    ws[1] = 0.0f;
}

// Streaming sum-of-squares over a float4-aligned array. Memory-bound: this is
// the 128MB entity table scan; also warms the 192MB L2 for the gather kernel.
__global__ __launch_bounds__(256) void l2_kernel(const float4* __restrict__ p,
                                                 int n4, float* __restrict__ acc) {
    float s = 0.0f;
    int stride = gridDim.x * blockDim.x;
    for (int i = blockIdx.x * blockDim.x + threadIdx.x; i < n4; i += stride) {
        if (i + stride < n4) __builtin_prefetch(&p[i + stride], 0, 1);
        float4 v = p[i];
        s = fmaf(v.x, v.x, fmaf(v.y, v.y, fmaf(v.z, v.z, fmaf(v.w, v.w, s))));
    }
    // wave32 reduction
    #pragma unroll
    for (int off = 16; off > 0; off >>= 1) s += __shfl_down(s, off, 32);
    if ((threadIdx.x & 31) == 0) atomicAdd(acc, s);
}

// ---------------------------------------------------------------------------
// Main KGCN kernel: one wave handles one tile of 16 batch rows.
//   - per-lane (lane pair) gather + attention + neighbor aggregation (f32)
//   - 16x16 GEMM vs agg_W^T via 4x V_WMMA_F32_16X16X4_F32 (K=16 total)
//   - tanh, LDS transpose, per-row logit, stable BCE, wave reduce, atomicAdd
// ---------------------------------------------------------------------------
__global__ __launch_bounds__(256) void kgcn_main_kernel(
    const float* __restrict__ user_emb,     // [N_USER, 16]
    const float* __restrict__ entity_emb,   // [N_ENTITY, 16]
    const float* __restrict__ relation_emb, // [N_RELATION, 16]
    const float* __restrict__ agg_W,        // [16, 16] row-major
    const float* __restrict__ agg_b,        // [16]
    const float* __restrict__ labels,       // [B]
    const int*   __restrict__ user_idx,     // [B]
    const int*   __restrict__ item_idx,     // [B]
    const int*   __restrict__ adj_e,        // [N_ENTITY, 8]
    const int*   __restrict__ adj_r,        // [N_ENTITY, 8]
    float* __restrict__ ws)
{
    __shared__ float lds_item[8][DIM * 16];   // 1KB per wave (item_emb tile)

    const int lane = threadIdx.x & 31;
    const int r    = lane & 15;       // row within tile (both halves mirror it)
    const int half = lane >> 4;       // 0: k/m 0..7 side, 1: k/m 2,3 / 8..15 side
    const int wave = threadIdx.x >> 5;
    const int tile = blockIdx.x * 8 + wave;   // 4096 tiles exactly
    const int row  = tile * 16 + r;

    const int   ui  = user_idx[row];
    const int   ii  = item_idx[row];
    const float lab = labels[row];

    // ---- gather u and self vectors (f32, float4 vectorized) ----
    float u[DIM], x[DIM];
    {
        const float4* up = (const float4*)(user_emb   + (long)ui * DIM);
        const float4* sp = (const float4*)(entity_emb + (long)ii * DIM);
        #pragma unroll
        for (int q = 0; q < 4; ++q) {
            float4 uv = up[q], sv = sp[q];
            u[4*q+0] = uv.x; u[4*q+1] = uv.y; u[4*q+2] = uv.z; u[4*q+3] = uv.w;
            x[4*q+0] = sv.x; x[4*q+1] = sv.y; x[4*q+2] = sv.z; x[4*q+3] = sv.w;
        }
    }

    // ---- attention scores: mean(u * rel_vec) over DIM ----
    int   ek[NNBR];
    float sc[NNBR];
    #pragma unroll
    for (int k = 0; k < NNBR; ++k) {
        ek[k]  = adj_e[(long)ii * NNBR + k];
        int rl = adj_r[(long)ii * NNBR + k];
        const float4* rp = (const float4*)(relation_emb + (long)rl * DIM);
        float d = 0.0f;
        #pragma unroll
        for (int q = 0; q < 4; ++q) {
            float4 rv = rp[q];
            d = fmaf(u[4*q+0], rv.x, d);
            d = fmaf(u[4*q+1], rv.y, d);
            d = fmaf(u[4*q+2], rv.z, d);
            d = fmaf(u[4*q+3], rv.w, d);
        }
        sc[k] = d * (1.0f / DIM);
    }

    // ---- softmax over 8 neighbors ----
    float mx = sc[0];
    #pragma unroll
    for (int k = 1; k < NNBR; ++k) mx = fmaxf(mx, sc[k]);
    float den = 0.0f;
    #pragma unroll
    for (int k = 0; k < NNBR; ++k) { sc[k] = expf(sc[k] - mx); den += sc[k]; }
    float inv = 1.0f / den;
    #pragma unroll
    for (int k = 0; k < NNBR; ++k) sc[k] *= inv;

    // ---- x = self_vec + sum_k attn_k * nbr_vec_k ----
    #pragma unroll
    for (int k = 0; k < NNBR; ++k) {
        const float4* np = (const float4*)(entity_emb + (long)ek[k] * DIM);
        #pragma unroll
        for (int q = 0; q < 4; ++q) {
            float4 nv = np[q];
            x[4*q+0] = fmaf(sc[k], nv.x, x[4*q+0]);
            x[4*q+1] = fmaf(sc[k], nv.y, x[4*q+1]);
            x[4*q+2] = fmaf(sc[k], nv.z, x[4*q+2]);
            x[4*q+3] = fmaf(sc[k], nv.w, x[4*q+3]);
        }
    }

    // ---- Y = X(16x16) @ W^T + b via 4 chained V_WMMA_F32_16X16X4_F32 ----
    // A frag (16x4): lane half 0 -> k-local {0,1}, half 1 -> k-local {2,3}
    // B frag (4x16): element (k_local, N=r) = W^T[4c+k_local][r] = agg_W[r*16 + 4c + k_local]
    // C/D (16x16):   lane (half,r) VGPR j -> Y[j + 8*half][r]
    const float bias = agg_b[r];
    v8f acc;
    #pragma unroll
    for (int j = 0; j < 8; ++j) acc[j] = bias;

    #pragma unroll
    for (int c = 0; c < 4; ++c) {
        v2f a, bm;
        a.x = half ? x[4*c + 2] : x[4*c + 0];
        a.y = half ? x[4*c + 3] : x[4*c + 1];
        const float* wrow = agg_W + r * DIM + 4*c + 2*half;
        bm.x = wrow[0];
        bm.y = wrow[1];
        acc = __builtin_amdgcn_wmma_f32_16x16x4_f32(
            /*neg_a=*/false, a, /*neg_b=*/false, bm,
            /*c_mod=*/(short)0, acc, /*reuse_a=*/false, /*reuse_b=*/false);
    }

    // ---- tanh + transpose tile through LDS ----
    #pragma unroll
    for (int j = 0; j < 8; ++j)
        lds_item[wave][(j + 8*half) * DIM + r] = tanhf(acc[j]);
    __syncthreads();

    // ---- logit + stable BCE: loss = softplus(logit) - label*logit ----
    float contrib = 0.0f;
    if (half == 0) {
        float logit = 0.0f;
        #pragma unroll
        for (int d = 0; d < DIM; ++d)
            logit = fmaf(u[d], lds_item[wave][r * DIM + d], logit);
        float sp = fmaxf(logit, 0.0f) + log1pf(expf(-fabsf(logit)));
        contrib = sp - lab * logit;
    }
    #pragma unroll
    for (int off = 16; off > 0; off >>= 1) contrib += __shfl_down(contrib, off, 32);
    if (lane == 0) atomicAdd(&ws[0], contrib);
}

__global__ void finalize_kernel(const float* __restrict__ ws, float* __restrict__ out) {
    // mean BCE + L2_WEIGHT * 0.5 * sum_sq
    out[0] = ws[0] * (1.0f / (float)BATCH) + 1e-4f * 0.5f * ws[1];
}

// ---------------------------------------------------------------------------
extern "C" void kernel_launch(void* const* d_in, const int* in_sizes, int n_in,
                              void* d_out, int out_size, void* d_ws, size_t ws_size,
                              hipStream_t stream) {
    const float* user_emb     = (const float*)d_in[0];
    const float* entity_emb   = (const float*)d_in[1];
    const float* relation_emb = (const float*)d_in[2];
    const float* agg_W        = (const float*)d_in[3];
    const float* agg_b        = (const float*)d_in[4];
    const float* labels       = (const float*)d_in[5];
    const int*   user_idx     = (const int*)d_in[6];
    const int*   item_idx     = (const int*)d_in[7];
    const int*   adj_e        = (const int*)d_in[8];
    const int*   adj_r        = (const int*)d_in[9];
    float* ws  = (float*)d_ws;
    float* out = (float*)d_out;

    zero_ws_kernel<<<1, 1, 0, stream>>>(ws);

    // Entity table first: 128MB stream fits in the 192MB L2, warming it for
    // the gather-heavy main kernel that follows on the same stream.
    l2_kernel<<<2048, 256, 0, stream>>>((const float4*)entity_emb,   in_sizes[1] / 4, ws + 1);
    l2_kernel<<< 256, 256, 0, stream>>>((const float4*)user_emb,     in_sizes[0] / 4, ws + 1);
    l2_kernel<<<   1, 256, 0, stream>>>((const float4*)relation_emb, in_sizes[2] / 4, ws + 1);

    // 65536 rows / 16 per tile / 8 waves per block = 512 blocks
    kgcn_main_kernel<<<BATCH / 16 / 8, 256, 0, stream>>>(
        user_emb, entity_emb, relation_emb, agg_W, agg_b, labels,
        user_idx, item_idx, adj_e, adj_r, ws);

    finalize_kernel<<<1, 1, 0, stream>>>(ws, out);
}